// InpaintContextAttentionUnit_19052474925472
// MI455X (gfx1250) — compile-verified
//
#include <hip/hip_runtime.h>
#include <hip/hip_bf16.h>

// MI455X (gfx1250, wave32). Memory-bound: ~536MB HBM traffic floor (~23us at
// 23.3 TB/s) if fm (134MB) stays resident in the 192MB L2 between kernels —
// hence NT stores for the 402MB write-once output. The 16ch->16feat masked
// conv maps exactly onto V_WMMA_F32_16X16X4_F32.

typedef float v2f __attribute__((ext_vector_type(2)));
typedef float v4f __attribute__((ext_vector_type(4)));
typedef float v8f __attribute__((ext_vector_type(8)));

#define B_   8
#define H_   512
#define W_   512
#define C_   16
#define NP_  8      // pooled height
#define WP_  256    // pooled width

__device__ __forceinline__ int iclamp(int v, int lo, int hi) {
    return v < lo ? lo : (v > hi ? hi : v);
}

__device__ __forceinline__ v4f shfl_xor_v4(v4f v, int mask) {
    v4f r;
    r.x = __shfl_xor(v.x, mask, 32);
    r.y = __shfl_xor(v.y, mask, 32);
    r.z = __shfl_xor(v.z, mask, 32);
    r.w = __shfl_xor(v.w, mask, 32);
    return r;
}

// ---------------------------------------------------------------------------
// Kernel 1: AvgPool (64,2): fm[B,512,512,16] -> pooled[B,8,256,16]
// One wave per pooled cell. The cell's data = 64 rows x 128B (2 pixels x 16ch).
// lane = rq*8 + q: rq = row-within-quad (0..3), q = float4 chunk (0..7) of the
// 32-float row segment. 16 iterations of global_load_b128 per lane; wave moves
// 512B per iteration. Reduce rows with xor16+xor8, w-pair with xor4.
// ---------------------------------------------------------------------------
__global__ __launch_bounds__(256) void pool_kernel(
    const float* __restrict__ fm, float* __restrict__ pooled)
{
    const int wid  = blockIdx.x * 8 + (threadIdx.x >> 5);   // 0..16383
    const int lane = threadIdx.x & 31;
    const int q  = lane & 7;    // float4 chunk: q<4 -> w=2j, c=4q; q>=4 -> w=2j+1
    const int rq = lane >> 3;   // row offset within quad of rows

    const int j = wid & 255;          // pooled w
    const int i = (wid >> 8) & 7;     // pooled h
    const int b = wid >> 11;

    const float* base =
        fm + (((long)b * H_ + i * 64) * W_ + 2 * j) * C_ + q * 4;

    v4f acc = {0.f, 0.f, 0.f, 0.f};
#pragma unroll 4
    for (int h0 = 0; h0 < 64; h0 += 4) {
        const v4f* p = (const v4f*)(base + (long)(h0 + rq) * (W_ * C_));
        v4f x = *p;
        acc.x += x.x; acc.y += x.y; acc.z += x.z; acc.w += x.w;
    }

    // combine rows (rq bits) then the two pooled columns (q bit 2)
    v4f t;
    t = shfl_xor_v4(acc, 16); acc.x += t.x; acc.y += t.y; acc.z += t.z; acc.w += t.w;
    t = shfl_xor_v4(acc,  8); acc.x += t.x; acc.y += t.y; acc.z += t.z; acc.w += t.w;
    t = shfl_xor_v4(acc,  4); acc.x += t.x; acc.y += t.y; acc.z += t.z; acc.w += t.w;

    if (lane < 4) {
        acc.x *= (1.0f / 128.0f); acc.y *= (1.0f / 128.0f);
        acc.z *= (1.0f / 128.0f); acc.w *= (1.0f / 128.0f);
        *(v4f*)(pooled + (long)wid * C_ + lane * 4) = acc;
    }
}

// ---------------------------------------------------------------------------
// Kernel 2: masked 3x3 conv (+bias, ReLU) via V_WMMA_F32_16X16X4_F32.
// mode 0: row conv on pooled[B,8,256,16]
// mode 1: col conv (conv of transposed pooled, written back untransposed):
//         col[b,i,j,f] = sum_{kh in {0,2}, kw} pooled[b,i+kw-1,j+kh-1,c]
//                        * kernel[kh,kw,c,f]
// One wave computes a 16(positions along j) x 16(features) tile.
// A: 16x4 f32 (pos x chan), B: 4x16 f32 (chan x feat), C/D: 16x16 f32.
// Layouts (ISA 7.12.2): A/B lane holds K = v + 2*(lane/16); D VGPR r holds
// M = r + 8*(lane/16), N = lane%16. kh==1 never visited -> the row mask.
// ---------------------------------------------------------------------------
__global__ __launch_bounds__(256) void conv_wmma_kernel(
    const float* __restrict__ pooled, const float* __restrict__ kern,
    const float* __restrict__ bias, float* __restrict__ rowb,
    float* __restrict__ colb)
{
    const int wid  = blockIdx.x * 8 + (threadIdx.x >> 5);   // 0..2047
    const int lane = threadIdx.x & 31;
    const int ln16 = lane & 15;
    const int half = lane >> 4;

    const int mode = wid >> 10;       // 0=row, 1=col
    const int rem  = wid & 1023;
    const int b    = rem >> 7;        // 0..7
    const int i    = (rem >> 4) & 7;  // 0..7
    const int j0   = (rem & 15) << 4; // 0..240 step 16

    // C initialized with bias (N = feature = ln16), accumulated through WMMA.
    v8f acc;
    const float bv = bias[ln16];
#pragma unroll
    for (int r = 0; r < 8; ++r) acc[r] = bv;

#pragma unroll
    for (int kh = 0; kh <= 2; kh += 2) {
#pragma unroll
        for (int kw = 0; kw < 3; ++kw) {
            const int oi = (mode == 0) ? (kh - 1) : (kw - 1);
            const int oj = (mode == 0) ? (kw - 1) : (kh - 1);
            const int ii = i + oi;
            const int jj = j0 + ln16 + oj;                 // A-row M = ln16
            const bool inb = ((unsigned)ii < (unsigned)NP_) &&
                             ((unsigned)jj < (unsigned)WP_);
            const float* aptr =
                pooled + (((long)(b * NP_ + ii) * WP_ + jj) * C_);
            const int tap = (kh * 3 + kw) * (C_ * C_);      // kernel[kh][kw]
#pragma unroll
            for (int k0 = 0; k0 < C_; k0 += 4) {
                const int kc = k0 + 2 * half;              // K = v + 2*(l/16)
                v2f a;
                if (inb) { a.x = aptr[kc]; a.y = aptr[kc + 1]; }
                else     { a.x = 0.f;      a.y = 0.f; }
                v2f wv;                                    // B[K][N=ln16]
                wv.x = kern[tap + kc * C_ + ln16];
                wv.y = kern[tap + (kc + 1) * C_ + ln16];
                acc = __builtin_amdgcn_wmma_f32_16x16x4_f32(
                    false, a, false, wv, (short)0, acc, false, false);
            }
        }
    }

    float* outp = (mode == 0) ? rowb : colb;
#pragma unroll
    for (int r = 0; r < 8; ++r) {                          // D: M = r + 8*half
        float v = acc[r] > 0.f ? acc[r] : 0.f;             // ReLU
        const int M = r + 8 * half;
        outp[((long)(b * NP_ + i) * WP_ + j0 + M) * C_ + ln16] = v;
    }
}

// ---------------------------------------------------------------------------
// Kernel 3: bilinear upsample (half-pixel centers, edge clamp) of row/col maps
// to 512x512, fuse: out = [fm, fm - row_up, fm - col_up].
// One thread per (pixel, 4-channel group): every access is a 16B-aligned
// float4 (pixel stride 48 ch = 192B). fm read is a contiguous 512B/wave
// b128 stream (L2 hit if fm survived since kernel 1). Output stores are
// non-temporal so the 402MB write-once stream doesn't evict fm from L2.
// ---------------------------------------------------------------------------
__global__ __launch_bounds__(256) void fuse_kernel(
    const float* __restrict__ fm, const float* __restrict__ rowb,
    const float* __restrict__ colb, float* __restrict__ out)
{
    const long t = (long)blockIdx.x * 256 + threadIdx.x;   // B*H*W*4 threads
    const int c = (int)(t & 3) * 4;                        // channel group
    const long p = t >> 2;                                 // pixel index
    const int w = (int)(p & 511);
    const int h = (int)((p >> 9) & 511);
    const int b = (int)(p >> 18);

    const v4f fmv = *(const v4f*)(fm + p * C_ + c);

    // jax.image.resize bilinear: src = (dst + 0.5) * (in/out) - 0.5, clamped.
    const float fy = (h + 0.5f) * (1.0f / 64.0f) - 0.5f;   // 8   -> 512
    const float fx = (w + 0.5f) * 0.5f - 0.5f;             // 256 -> 512
    const int y0 = (int)floorf(fy);
    const int x0 = (int)floorf(fx);
    const float wy = fy - (float)y0;
    const float wx = fx - (float)x0;
    const int y0c = iclamp(y0, 0, NP_ - 1), y1c = iclamp(y0 + 1, 0, NP_ - 1);
    const int x0c = iclamp(x0, 0, WP_ - 1), x1c = iclamp(x0 + 1, 0, WP_ - 1);

    const long base = (long)b * NP_ * WP_ * C_ + c;
    const long i00 = base + ((long)y0c * WP_ + x0c) * C_;
    const long i01 = base + ((long)y0c * WP_ + x1c) * C_;
    const long i10 = base + ((long)y1c * WP_ + x0c) * C_;
    const long i11 = base + ((long)y1c * WP_ + x1c) * C_;

    const float w00 = (1.f - wy) * (1.f - wx);
    const float w01 = (1.f - wy) * wx;
    const float w10 = wy * (1.f - wx);
    const float w11 = wy * wx;

    const v4f r00 = *(const v4f*)(rowb + i00), r01 = *(const v4f*)(rowb + i01);
    const v4f r10 = *(const v4f*)(rowb + i10), r11 = *(const v4f*)(rowb + i11);
    const v4f c00 = *(const v4f*)(colb + i00), c01 = *(const v4f*)(colb + i01);
    const v4f c10 = *(const v4f*)(colb + i10), c11 = *(const v4f*)(colb + i11);

    v4f rv, cv, d0, d1;
#pragma unroll
    for (int k = 0; k < 4; ++k) {
        rv[k] = w00 * r00[k] + w01 * r01[k] + w10 * r10[k] + w11 * r11[k];
        cv[k] = w00 * c00[k] + w01 * c01[k] + w10 * c10[k] + w11 * c11[k];
        d0[k] = fmv[k] - rv[k];
        d1[k] = fmv[k] - cv[k];
    }

    float* o = out + p * 48 + c;
    __builtin_nontemporal_store(fmv, (v4f*)(o));
    __builtin_nontemporal_store(d0,  (v4f*)(o + 16));
    __builtin_nontemporal_store(d1,  (v4f*)(o + 32));
}

// ---------------------------------------------------------------------------
extern "C" void kernel_launch(void* const* d_in, const int* in_sizes, int n_in,
                              void* d_out, int out_size, void* d_ws, size_t ws_size,
                              hipStream_t stream)
{
    const float* fm   = (const float*)d_in[0];   // [8,512,512,16]
    const float* kern = (const float*)d_in[1];   // [3,3,16,16]
    const float* bias = (const float*)d_in[2];   // [16]
    float* out = (float*)d_out;                  // [8,512,512,48]

    const long POOL_N = (long)B_ * NP_ * WP_ * C_;   // 262144 floats
    float* pooled = (float*)d_ws;
    float* rowb   = pooled + POOL_N;
    float* colb   = rowb + POOL_N;                   // total 3MB of d_ws

    // 1) AvgPool: 16384 waves, 8 waves/block, b128 loads
    pool_kernel<<<16384 / 8, 256, 0, stream>>>(fm, pooled);

    // 2) WMMA conv (row + col modes): 2048 waves, 8 waves/block
    conv_wmma_kernel<<<2048 / 8, 256, 0, stream>>>(pooled, kern, bias, rowb, colb);

    // 3) Upsample + diff + concat: B*H*W*4 threads, all-b128, NT stores
    const long total = (long)B_ * H_ * W_ * 4;       // 8,388,608
    fuse_kernel<<<(unsigned)(total / 256), 256, 0, stream>>>(fm, rowb, colb, out);
}